// GAT_16011638079940
// MI455X (gfx1250) — compile-verified
//
#include <hip/hip_runtime.h>

typedef float v2f __attribute__((ext_vector_type(2)));
typedef float v8f __attribute__((ext_vector_type(8)));
typedef unsigned int u32x4 __attribute__((ext_vector_type(4)));
typedef int i32x4 __attribute__((ext_vector_type(4)));
typedef int i32x8 __attribute__((ext_vector_type(8)));

#define FW 128  // feature width of every propagated tensor (F_IN == HID == 128)

// ---------------- zero fill (grid-stride) ----------------
__global__ void zero_f32(float* __restrict__ p, long n) {
  long i = (long)blockIdx.x * blockDim.x + threadIdx.x;
  long stride = (long)gridDim.x * blockDim.x;
  for (; i < n; i += stride) p[i] = 0.0f;
}

// ---------------- degree: deg[dst[e]] += 1 ----------------
__global__ void deg_kernel(const int* __restrict__ dst, float* __restrict__ deg, int E) {
  int e = blockIdx.x * blockDim.x + threadIdx.x;
  if (e < E) atomicAdd(&deg[dst[e]], 1.0f);
}

// ---------------- dis = deg>0 ? deg^-0.5 : 0 (in place) ----------------
__global__ void dis_kernel(float* __restrict__ deg, int N) {
  int i = blockIdx.x * blockDim.x + threadIdx.x;
  if (i < N) {
    float d = deg[i];
    deg[i] = (d > 0.0f) ? rsqrtf(d) : 0.0f;
  }
}

// ---------------- one hop: hn[dst] += dis[src]*dis[dst]*h[src] ----------------
// one wave32 per edge; each lane moves one float4 (128 floats per row).
// Working set (~77MB) is resident in the 192MB L2, so the no-return f32
// atomics (global_atomic_add_f32, STOREcnt-tracked) resolve in L2, not HBM.
__global__ void __launch_bounds__(256) prop_kernel(
    const float* __restrict__ h, const int* __restrict__ src,
    const int* __restrict__ dst, const float* __restrict__ dis,
    float* __restrict__ hn, int E) {
  int gid = blockIdx.x * blockDim.x + threadIdx.x;
  int e = gid >> 5;
  int lane = gid & 31;
  if (e >= E) return;
  int s = src[e];
  int d = dst[e];
  float w = dis[s] * dis[d];
  const float4* hrow = (const float4*)(h + (size_t)s * FW);
  float4 v = hrow[lane];
  float* orow = hn + (size_t)d * FW + lane * 4;
  atomicAdd(orow + 0, w * v.x);
  atomicAdd(orow + 1, w * v.y);
  atomicAdd(orow + 2, w * v.z);
  atomicAdd(orow + 3, w * v.w);
}

// ---------------- WMMA fp32 GEMM: C(MxNc) (+)= A(Mx128) * B(128xNc) ----------------
// 8 waves / block. Wave 0 DMAs the full 128xNc B panel into LDS with the
// Tensor Data Mover (tensor_load_to_lds + s_wait_tensorcnt), then each wave
// owns one 16-row A strip and sweeps all Nc/16 column tiles: one v2f A load
// feeds NT WMMAs per K-step, B comes from ds_load. Accumulation across the 4
// hop GEMMs goes through memory (acc flag); bias/ReLU fused in the last pass.
template <int NT>
__global__ void __launch_bounds__(256) wmma_gemm(
    const float* __restrict__ A, const float* __restrict__ B,
    const float* __restrict__ bias, float* __restrict__ Cm,
    int M, int accFlag, int reluFlag) {
  constexpr int Nc = NT * 16;
  __shared__ float bs[FW * Nc];  // 128 x Nc B panel (row-major, packed)

  // ---- TDM: stage B panel into LDS (wave 0 issues, all waves barrier) ----
  if (threadIdx.x < 32) {
    unsigned long long gaddr = (unsigned long long)(const void*)B;
    unsigned int laddr = (unsigned int)(unsigned long long)(void*)bs;
    u32x4 g0 = {};
    g0[0] = 1u;                                   // count=1 (valid descriptor)
    g0[1] = laddr;                                // lds_addr (bytes)
    g0[2] = (unsigned int)gaddr;                  // global_addr[31:0]
    g0[3] = (unsigned int)((gaddr >> 32) & 0x1FFFFFFu) | (2u << 30); // addr[56:32] | type=2
    i32x8 g1 = {};
    g1[0] = (int)(2u << 16);                      // data_size=2 (4B), mask=0
    g1[1] = (int)(((unsigned)Nc & 0xFFFFu) << 16);          // tensor_dim0 lo16
    g1[2] = (int)((((unsigned)Nc >> 16) & 0xFFFFu) |
                  (((unsigned)FW & 0xFFFFu) << 16));        // td0 hi | tensor_dim1 lo16
    g1[3] = (int)(((((unsigned)FW >> 16) & 0xFFFFu)) |
                  (((unsigned)Nc) << 16));                  // td1 hi | tile_dim0=Nc
    g1[4] = (int)((unsigned)FW);                  // tile_dim1=128, tile_dim2=0
    g1[5] = (int)((unsigned)Nc);                  // tensor_dim0_stride lo32
    g1[6] = 0;                                    // stride hi | dim1_stride lo
    g1[7] = 0;
    i32x4 g2 = {};                                // 2-D tensor: groups 2/3 zero
    i32x4 g3 = {};
    i32x8 g4 = {};                                // extra operand of 6-arg form
    __builtin_amdgcn_tensor_load_to_lds(g0, g1, g2, g3, g4, 0);
    __builtin_amdgcn_s_wait_tensorcnt(0);
  }
  __syncthreads();

  int wave = threadIdx.x >> 5;
  int lane = threadIdx.x & 31;   // EXEC all-ones inside the WMMA loop
  int half = lane >> 4;          // 0: K pair {0,1}; 1: K pair {2,3}
  int l = lane & 15;             // A row / B,C column within the tile

  int tm = blockIdx.x * 8 + wave;
  if (tm * 16 >= M) return;      // after the only barrier: safe to exit
  int m0 = tm * 16;

  v8f acc[NT];
#pragma unroll
  for (int nt = 0; nt < NT; ++nt) acc[nt] = (v8f){};
  if (accFlag) {
#pragma unroll
    for (int nt = 0; nt < NT; ++nt)
#pragma unroll
      for (int r = 0; r < 8; ++r)
        acc[nt][r] = Cm[(size_t)(m0 + r + half * 8) * Nc + nt * 16 + l];
  }

  // A layout (ISA 7.12.2, 32-bit A 16x4): lanes L and L+16 both hold row L;
  // VGPRs hold K={0,1} (lanes 0-15) or K={2,3} (lanes 16-31).
  const float* aptr = A + (size_t)(m0 + l) * FW + half * 2;

#pragma unroll 2
  for (int k = 0; k < FW; k += 4) {
    v2f a = *(const v2f*)(aptr + k);
    int kb = k + half * 2;
    const float* brow0 = bs + kb * Nc + l;
#pragma unroll
    for (int nt = 0; nt < NT; ++nt) {
      v2f b;
      b.x = brow0[nt * 16];         // B[kb][nt*16+l]
      b.y = brow0[Nc + nt * 16];    // B[kb+1][nt*16+l]
      acc[nt] = __builtin_amdgcn_wmma_f32_16x16x4_f32(
          false, a, false, b, (short)0, acc[nt], false, false);
    }
  }

#pragma unroll
  for (int nt = 0; nt < NT; ++nt) {
    float bv = bias ? bias[nt * 16 + l] : 0.0f;
#pragma unroll
    for (int r = 0; r < 8; ++r) {
      float v = acc[nt][r] + bv;
      if (reluFlag) v = fmaxf(v, 0.0f);
      Cm[(size_t)(m0 + r + half * 8) * Nc + nt * 16 + l] = v;
    }
  }
}

extern "C" void kernel_launch(void* const* d_in, const int* in_sizes, int n_in,
                              void* d_out, int out_size, void* d_ws, size_t ws_size,
                              hipStream_t stream) {
  const float* x  = (const float*)d_in[0];
  const int*   ei = (const int*)d_in[1];
  const float* w1 = (const float*)d_in[2];
  const float* b1 = (const float*)d_in[3];
  const float* w2 = (const float*)d_in[4];
  const float* b2 = (const float*)d_in[5];
  float* out = (float*)d_out;

  const int N   = in_sizes[0] / FW;   // 50000 (divisible by 16)
  const int E   = in_sizes[1] / 2;    // 800000
  const int HID = in_sizes[3];        // 128
  const int C   = in_sizes[5];        // 32

  const int* src = ei;
  const int* dst = ei + E;

  // workspace layout: dis[N] | hA[N*128] | hB[N*128] | hid[N*128]
  float* dis = (float*)d_ws;
  size_t off = (((size_t)N * sizeof(float)) + 255) & ~(size_t)255;
  float* hA  = (float*)((char*)d_ws + off); off += (size_t)N * FW * sizeof(float);
  float* hB  = (float*)((char*)d_ws + off); off += (size_t)N * FW * sizeof(float);
  float* hid = (float*)((char*)d_ws + off);

  const long NF = (long)N * FW;
  const int propBlocks = (int)(((long)E * 32 + 255) / 256);
  const int Mtiles = N / 16;                // 3125
  const int gemmBlocks = (Mtiles + 7) / 8;  // 8 waves (tiles) per block

  // ---- symmetric normalization weights ----
  zero_f32<<<2048, 256, 0, stream>>>(dis, N);
  deg_kernel<<<(E + 255) / 256, 256, 0, stream>>>(dst, dis, E);
  dis_kernel<<<(N + 255) / 256, 256, 0, stream>>>(dis, N);

  // ---- layer 1: hid = relu(sum_k hk @ w1[k] + b1) ----
  wmma_gemm<8><<<gemmBlocks, 256, 0, stream>>>(x, w1 + 0 * FW * HID, nullptr, hid, N, 0, 0);
  zero_f32<<<4096, 256, 0, stream>>>(hA, NF);
  prop_kernel<<<propBlocks, 256, 0, stream>>>(x, src, dst, dis, hA, E);
  wmma_gemm<8><<<gemmBlocks, 256, 0, stream>>>(hA, w1 + 1 * FW * HID, nullptr, hid, N, 1, 0);
  zero_f32<<<4096, 256, 0, stream>>>(hB, NF);
  prop_kernel<<<propBlocks, 256, 0, stream>>>(hA, src, dst, dis, hB, E);
  wmma_gemm<8><<<gemmBlocks, 256, 0, stream>>>(hB, w1 + 2 * FW * HID, nullptr, hid, N, 1, 0);
  zero_f32<<<4096, 256, 0, stream>>>(hA, NF);
  prop_kernel<<<propBlocks, 256, 0, stream>>>(hB, src, dst, dis, hA, E);
  wmma_gemm<8><<<gemmBlocks, 256, 0, stream>>>(hA, w1 + 3 * FW * HID, b1, hid, N, 1, 1);

  // ---- layer 2: out = sum_k gk @ w2[k] + b2 ----
  wmma_gemm<2><<<gemmBlocks, 256, 0, stream>>>(hid, w2 + 0 * HID * C, nullptr, out, N, 0, 0);
  zero_f32<<<4096, 256, 0, stream>>>(hA, NF);
  prop_kernel<<<propBlocks, 256, 0, stream>>>(hid, src, dst, dis, hA, E);
  wmma_gemm<2><<<gemmBlocks, 256, 0, stream>>>(hA, w2 + 1 * HID * C, nullptr, out, N, 1, 0);
  zero_f32<<<4096, 256, 0, stream>>>(hB, NF);
  prop_kernel<<<propBlocks, 256, 0, stream>>>(hA, src, dst, dis, hB, E);
  wmma_gemm<2><<<gemmBlocks, 256, 0, stream>>>(hB, w2 + 2 * HID * C, nullptr, out, N, 1, 0);
  zero_f32<<<4096, 256, 0, stream>>>(hA, NF);
  prop_kernel<<<propBlocks, 256, 0, stream>>>(hB, src, dst, dis, hA, E);
  wmma_gemm<2><<<gemmBlocks, 256, 0, stream>>>(hA, w2 + 3 * HID * C, b2, out, N, 1, 0);
}